// CombinedModel_sink_70437463655086
// MI455X (gfx1250) — compile-verified
//
#include <hip/hip_runtime.h>
#include <hip/hip_bf16.h>

#define Bn   64
#define NM   333
#define DD   512
#define TT   28
#define ITERS 100

typedef __attribute__((ext_vector_type(2))) float v2f;
typedef __attribute__((ext_vector_type(8))) float v8f;

// ---------------------------------------------------------------------------
// K1: TA[m,d] = sum_e text1[m,e] * A[d,e]   (both operands are contiguous rows)
// ---------------------------------------------------------------------------
__global__ __launch_bounds__(512)
void ta_kernel(const float* __restrict__ text1, const float* __restrict__ A,
               float* __restrict__ TA) {
  __shared__ float trow[DD];
  const int m = blockIdx.x;
  const int d = threadIdx.x;
  trow[d] = text1[(size_t)m * DD + d];
  __syncthreads();
  const float* ar = A + (size_t)d * DD;
  float acc = 0.f;
#pragma unroll 8
  for (int e = 0; e < DD; ++e) acc = fmaf(trow[e], ar[e], acc);
  TA[(size_t)m * DD + d] = acc;
}

// ---------------------------------------------------------------------------
// K2: S[b,n,m] = f[b,n,:] . TA[m,:]  via V_WMMA_F32_16X16X4_F32
//     one 16x16 output tile per wave, K looped in steps of 4.
// ---------------------------------------------------------------------------
__global__ __launch_bounds__(256)
void affinity_wmma_kernel(const float* __restrict__ feat,
                          const float* __restrict__ TA,
                          float* __restrict__ S) {
  const int b    = blockIdx.x;
  const int nt   = blockIdx.y;          // 0..20
  const int wave = threadIdx.x >> 5;    // 0..7
  const int lane = threadIdx.x & 31;
  const int mt   = blockIdx.z * 8 + wave;
  if (mt >= 21) return;                 // wave-uniform exit, EXEC stays full

  const int n0 = nt * 16, m0 = mt * 16;
  int ra = n0 + (lane & 15); if (ra > NM - 1) ra = NM - 1;   // clamp edge rows
  int rb = m0 + (lane & 15); if (rb > NM - 1) rb = NM - 1;
  const int kh = (lane >> 4) * 2;       // K-pair select per ISA A-layout

  const float* pa = feat + ((size_t)b * NM + ra) * DD + kh;
  const float* pb = TA + (size_t)rb * DD + kh;

  v8f acc = {};
#pragma unroll 4
  for (int k = 0; k < DD; k += 4) {
    v2f av = *(const v2f*)(pa);
    v2f bv = *(const v2f*)(pb);
    acc = __builtin_amdgcn_wmma_f32_16x16x4_f32(false, av, false, bv,
                                                (short)0, acc, false, false);
    pa += 4; pb += 4;
  }

  const int col   = m0 + (lane & 15);
  const int rbase = n0 + ((lane >> 4) << 3);
  float* out = S + (size_t)b * NM * NM;
#pragma unroll
  for (int v = 0; v < 8; ++v) {
    const int row = rbase + v;
    if (row < NM && col < NM) out[(size_t)row * NM + col] = acc[v];
  }
}

// ---------------------------------------------------------------------------
// K3: fused InstanceNorm + 100 Sinkhorn iterations + row-sums of exp(la).
//     One 11-wave workgroup per batch; tensor is L2 resident.
//     Each iteration = a SINGLE read+write pass over the matrix: the previous
//     iteration's column lse is subtracted on load, the row logsumexp is done
//     in registers, and the new column masses are accumulated in registers
//     (lane <-> column mapping is fixed), merged via LDS once per iteration.
// ---------------------------------------------------------------------------
__global__ __launch_bounds__(352)
void sinkhorn_kernel(float* __restrict__ la, float* __restrict__ rowsum,
                     const float* __restrict__ gamma,
                     const float* __restrict__ beta) {
  const int b    = blockIdx.x;
  const int tid  = threadIdx.x;
  const int lane = tid & 31;
  const int wave = tid >> 5;            // 0..10
  float* L = la + (size_t)b * NM * NM;

  __shared__ float part[11 * NM];       // per-wave column mass partials
  __shared__ float col_lse[NM];
  __shared__ float red[22];
  __shared__ float bc[2];

  // ---- InstanceNorm statistics (population var, eps = 1e-5) ----
  float sum = 0.f, sq = 0.f;
  for (int i = tid; i < NM * NM; i += 352) {
    const float v = L[i];
    sum += v; sq += v * v;
  }
  for (int o = 16; o; o >>= 1) {
    sum += __shfl_xor(sum, o, 32);
    sq  += __shfl_xor(sq,  o, 32);
  }
  if (lane == 0) { red[wave] = sum; red[11 + wave] = sq; }
  __syncthreads();
  if (tid == 0) {
    float ts = 0.f, tq = 0.f;
    for (int w = 0; w < 11; ++w) { ts += red[w]; tq += red[11 + w]; }
    const float inv = 1.0f / (float)(NM * NM);
    const float mu  = ts * inv;
    const float var = tq * inv - mu * mu;
    const float rs  = rsqrtf(var + 1e-5f);
    bc[0] = rs * gamma[0];
    bc[1] = beta[0] - mu * rs * gamma[0];
  }
  for (int m = tid; m < NM; m += 352) col_lse[m] = 0.f;  // iter-0: no col norm yet
  __syncthreads();
  const float nscl = bc[0], noff = bc[1];

  for (int it = 0; it < ITERS; ++it) {
    const float a = (it == 0) ? nscl : 1.0f;   // fold InstanceNorm into iter 0
    const float c = (it == 0) ? noff : 0.0f;

    // cache previous column lse + zero column-mass accumulators (registers)
    float cl[11], csum[11];
#pragma unroll
    for (int j = 0; j < 11; ++j) {
      const int m = lane + 32 * j;
      cl[j]   = (m < NM) ? col_lse[m] : 0.f;
      csum[j] = 0.f;
    }

    // fused pass: col-subtract (prev iter) + row normalization + col accumulate
    for (int n = wave; n < NM; n += 11) {
      float* rp = L + (size_t)n * NM;
      float v[11];
      float mx = -__builtin_inff();
#pragma unroll
      for (int j = 0; j < 11; ++j) {
        const int m = lane + 32 * j;
        v[j] = (m < NM) ? (fmaf(rp[m], a, c) - cl[j]) : -__builtin_inff();
        mx = fmaxf(mx, v[j]);
      }
      for (int o = 16; o; o >>= 1) mx = fmaxf(mx, __shfl_xor(mx, o, 32));
      float e[11];
      float se = 0.f;
#pragma unroll
      for (int j = 0; j < 11; ++j) { e[j] = __expf(v[j] - mx); se += e[j]; }
      for (int o = 16; o; o >>= 1) se += __shfl_xor(se, o, 32);
      const float lse = mx + __logf(se);
      const float inv = 1.0f / se;      // exp(v-lse) == e[j] * (1/se): reuse exps
#pragma unroll
      for (int j = 0; j < 11; ++j) {
        const int m = lane + 32 * j;
        if (m < NM) rp[m] = v[j] - lse;
        csum[j] = fmaf(e[j], inv, csum[j]);   // e[j]==0 for invalid m
      }
    }

    // merge per-wave column masses -> new column lse (no max shift needed:
    // all row-normalized entries are <= 0, so masses are in (0, NM])
#pragma unroll
    for (int j = 0; j < 11; ++j) {
      const int m = lane + 32 * j;
      if (m < NM) part[wave * NM + m] = csum[j];
    }
    __syncthreads();
    if (tid < NM) {
      float s = 0.f;
      for (int w = 0; w < 11; ++w) s += part[w * NM + tid];
      col_lse[tid] = __logf(s);
    }
    __syncthreads();
  }

  // ---- rowsum[b,n] = sum_m exp(r100 - col_lse[m])  (P never materialized;
  //      the last column subtraction is folded in via exp(-col_lse)) ----
  float ecl[11];
#pragma unroll
  for (int j = 0; j < 11; ++j) {
    const int m = lane + 32 * j;
    ecl[j] = (m < NM) ? __expf(-col_lse[m]) : 0.f;
  }
  for (int n = wave; n < NM; n += 11) {
    const float* rp = L + (size_t)n * NM;
    float s = 0.f;
#pragma unroll
    for (int j = 0; j < 11; ++j) {
      const int m = lane + 32 * j;
      if (m < NM) s += __expf(rp[m]) * ecl[j];
    }
    for (int o = 16; o; o >>= 1) s += __shfl_xor(s, o, 32);
    if (lane == 0) rowsum[b * NM + n] = s;
  }
}

// ---------------------------------------------------------------------------
// K4: mean_text[d] = (1/M) sum_m text1[m,d]
// ---------------------------------------------------------------------------
__global__ __launch_bounds__(512)
void meantext_kernel(const float* __restrict__ text1, float* __restrict__ mt) {
  const int d = threadIdx.x;
  float acc = 0.f;
  for (int m = 0; m < NM; ++m) acc += text1[(size_t)m * DD + d];
  mt[d] = acc * (1.0f / NM);
}

// ---------------------------------------------------------------------------
// K5: pooled1[b,d] = (1/M) sum_n f[b,n,d] * rowsum[b,n]
// ---------------------------------------------------------------------------
__global__ __launch_bounds__(512)
void pooled_kernel(const float* __restrict__ feat, const float* __restrict__ r,
                   float* __restrict__ pooled) {
  const int b = blockIdx.x;
  const int d = threadIdx.x;
  __shared__ float rs[NM];
  for (int i = threadIdx.x; i < NM; i += 512) rs[i] = r[b * NM + i];
  __syncthreads();
  const float* fb = feat + (size_t)b * NM * DD + d;
  float acc = 0.f;
  for (int n = 0; n < NM; ++n) acc = fmaf(fb[(size_t)n * DD], rs[n], acc);
  pooled[b * DD + d] = acc * (1.0f / NM);
}

// ---------------------------------------------------------------------------
// K6: prediction[b,t] = (pooled1.Whead[0:D] + mean_text.Whead[D:2D] + bh)*scale
//     plus the task_scale passthrough output.
// ---------------------------------------------------------------------------
__global__ __launch_bounds__(32)
void head_kernel(const float* __restrict__ pooled, const float* __restrict__ mt,
                 const float* __restrict__ W, const float* __restrict__ bh,
                 const float* __restrict__ ts, float* __restrict__ out) {
  const int b = blockIdx.x;
  const int t = threadIdx.x;
  if (t < TT) {
    float acc = bh[t];
    const float* pb = pooled + b * DD;
    for (int d = 0; d < DD; ++d) acc = fmaf(pb[d], W[(size_t)d * TT + t], acc);
    for (int d = 0; d < DD; ++d) acc = fmaf(mt[d], W[(size_t)(DD + d) * TT + t], acc);
    out[b * TT + t] = acc * ts[t];
    if (b == 0) out[Bn * TT + t] = ts[t];   // second tuple element
  }
}

// ---------------------------------------------------------------------------
extern "C" void kernel_launch(void* const* d_in, const int* in_sizes, int n_in,
                              void* d_out, int out_size, void* d_ws, size_t ws_size,
                              hipStream_t stream) {
  const float* feat  = (const float*)d_in[0];               // [64,333,512]
  const float* text  = (const float*)d_in[1];               // [2,333,512]
  const float* A     = (const float*)d_in[2];               // [512,512]
  const float* gamma = (const float*)d_in[3];
  const float* beta  = (const float*)d_in[4];
  const float* Wh    = (const float*)d_in[5];               // [1024,28]
  const float* bh    = (const float*)d_in[6];               // [28]
  const float* tsc   = (const float*)d_in[7];               // [28]
  float* out = (float*)d_out;

  const float* text1 = text + (size_t)NM * DD;              // text[1]

  float* ws = (float*)d_ws;
  float* TA     = ws;                                       // 333*512
  float* la     = TA + (size_t)NM * DD;                     // 64*333*333 (S in-place)
  float* rowsum = la + (size_t)Bn * NM * NM;                // 64*333
  float* mtext  = rowsum + (size_t)Bn * NM;                 // 512
  float* pooled = mtext + DD;                               // 64*512

  ta_kernel<<<NM, 512, 0, stream>>>(text1, A, TA);
  affinity_wmma_kernel<<<dim3(Bn, 21, 3), 256, 0, stream>>>(feat, TA, la);
  sinkhorn_kernel<<<Bn, 352, 0, stream>>>(la, rowsum, gamma, beta);
  meantext_kernel<<<1, 512, 0, stream>>>(text1, mtext);
  pooled_kernel<<<Bn, 512, 0, stream>>>(feat, rowsum, pooled);
  head_kernel<<<Bn, 32, 0, stream>>>(pooled, mtext, Wh, bh, tsc, out);
}